// FWBlock_36009005810046
// MI455X (gfx1250) — compile-verified
//
#include <hip/hip_runtime.h>
#include <math.h>

typedef __attribute__((ext_vector_type(2))) float v2f;
typedef __attribute__((ext_vector_type(8))) float v8f;

namespace {
constexpr int   Bn   = 8;
constexpr int   Cin  = 32;
constexpr int   Hh   = 96;
constexpr int   Ww   = 96;
constexpr int   Dd   = 128;            // DICT
constexpr int   HW   = Hh * Ww;        // 9216
constexpr long  Nt   = (long)Bn * Dd * HW;   // 9437184   (c / X tensors)
constexpr long  Nx   = (long)Bn * Cin * HW;  // 2359296   (x / xp tensors)
constexpr int   KD   = Cin * 9;        // 288  (fwd GEMM K)
constexpr int   KDT  = Dd * 9;         // 1152 (transpose GEMM K)
constexpr long  MMAX = Nt / 8;         // 1179648 max matricization columns
constexpr float TAUc = 100.0f;
}

#define DEV __device__ __forceinline__

DEV float wave_sum(float v) {
    for (int o = 16; o > 0; o >>= 1) v += __shfl_xor(v, o, 32);
    return v;
}

DEV int  mode_n(int m)      { return m == 0 ? 8 : (m == 1 ? 128 : 96); }
DEV long mode_stride(int m) { return m == 0 ? 1179648L : (m == 1 ? 9216L : (m == 2 ? 96L : 1L)); }

// column j of matricization mode m -> base element offset (stride = mode_stride)
DEV long col_base(int m, long j) {
    switch (m) {
    case 0: return j;
    case 1: { long i3 = j % 96, t = j / 96, i2 = t % 96,  i0 = t / 96;
              return i0 * 1179648L + i2 * 96 + i3; }
    case 2: { long i3 = j % 96, t = j / 96, i1 = t % 128, i0 = t / 128;
              return i0 * 1179648L + i1 * 9216L + i3; }
    default:{ long i2 = j % 96, t = j / 96, i1 = t % 128, i0 = t / 128;
              return i0 * 1179648L + i1 * 9216L + i2 * 96; }
    }
}

// element e -> (mode-m index i, matricization column j)
DEV void elem_ij(int m, long e, int& i, long& j) {
    int i3 = (int)(e % 96); long t = e / 96;
    int i2 = (int)(t % 96); t /= 96;
    int i1 = (int)(t % 128);
    int i0 = (int)(t / 128);
    switch (m) {
    case 0: i = i0; j = ((long)i1 * 96 + i2) * 96 + i3; break;
    case 1: i = i1; j = ((long)i0 * 96 + i2) * 96 + i3; break;
    case 2: i = i2; j = ((long)i0 * 128 + i1) * 96 + i3; break;
    default:i = i3; j = ((long)i0 * 128 + i1) * 96 + i2; break;
    }
}

// ---------------------------------------------------------------- utilities
__global__ void zero_k(float* p, long n) {
    long s = (long)gridDim.x * blockDim.x;
    for (long e = (long)blockIdx.x * blockDim.x + threadIdx.x; e < n; e += s) p[e] = 0.f;
}

__global__ void normalize_weight_k(const float* __restrict__ w, float* __restrict__ wn) {
    __shared__ float red[3];
    const int o = blockIdx.x, t = threadIdx.x;
    float s = 0.f;
    for (int k = t; k < KD; k += 96) { float v = w[o * KD + k]; s += v * v; }
    s = wave_sum(s);
    if ((t & 31) == 0) red[t >> 5] = s;
    __syncthreads();
    const float ss  = red[0] + red[1] + red[2];
    const float inv = 1.0f / fmaxf(sqrtf(ss), 1e-12f);
    for (int k = t; k < KD; k += 96) wn[o * KD + k] = w[o * KD + k] * inv;
}

__global__ void yita_k(const float* __restrict__ wn, float* __restrict__ scal) {
    __shared__ float red[8];
    float s = 0.f;
    for (int i = threadIdx.x; i < Dd * KD; i += 256) { float v = wn[i]; s += v * v; }
    s = wave_sum(s);
    if ((threadIdx.x & 31) == 0) red[threadIdx.x >> 5] = s;
    __syncthreads();
    if (threadIdx.x == 0) {
        float a = 0.f;
        for (int i = 0; i < 8; ++i) a += red[i];
        scal[0] = 8.0f * a;      // yita = yitaA * ||W||_F^2
    }
}

// transpose-conv weights, reordered once:  wtr[ci][k], k=o*9+ky*3+kx,
// value = wn[o][ci*9 + (2-ky)*3 + (2-kx)]  -> linear A-fragment loads later
__global__ void reorder_wt_k(const float* __restrict__ wn, float* __restrict__ wtr) {
    const int s = gridDim.x * blockDim.x;
    for (int idx = blockIdx.x * blockDim.x + threadIdx.x; idx < Cin * KDT; idx += s) {
        const int ci = idx / KDT, k = idx % KDT;
        const int o = k / 9, r9 = k % 9, ky = r9 / 3, kx = r9 % 3;
        wtr[idx] = wn[o * KD + ci * 9 + (2 - ky) * 3 + (2 - kx)];
    }
}

// ---------------------------------------------------------------- WMMA convs
// Forward conv as implicit GEMM: D(16 och x 16 pos) += A(16x4 wn) * B(4x16 im2col).
// One block = one 16-pixel tile; im2col slab staged to LDS once, shared by 8 waves
// (one 16-och tile each). Inner loop: global_load_b64(A) + ds_load_b64(B) + v_wmma.
__global__ void conv_fwd_wmma(const float* __restrict__ x, const float* __restrict__ wn,
                              const float* __restrict__ scal, float* __restrict__ Xc) {
    __shared__ float Bt[16 * KD];                // [col][k]  18KB
    const int ptile = blockIdx.x;                // 0..4607
    const long pos0 = (long)ptile * 16;
    const int  bt  = (int)(pos0 / HW);
    const int  rm  = (int)(pos0 % HW);
    const int  oy  = rm / Ww;
    const int  ox0 = rm % Ww;                    // multiple of 16: tile is one row chunk
    const float* xb = x + (long)bt * Cin * HW;

    for (int idx = threadIdx.x; idx < 16 * KD; idx += 256) {   // 18 iters exactly
        const int col = idx / KD;
        const int k   = idx % KD;
        const int ci = k / 9, r9 = k % 9, ky = r9 / 3, kx = r9 % 3;
        const int iy = oy + ky - 1, ix = ox0 + col + kx - 1;
        float v = 0.f;
        if (iy >= 0 && iy < Hh && ix >= 0 && ix < Ww) v = xb[(long)ci * HW + iy * Ww + ix];
        Bt[idx] = v;
    }
    __syncthreads();

    const int lane = threadIdx.x & 31;
    const int wv   = threadIdx.x >> 5;
    const int nl   = lane & 15;                  // A-row / B-col / D-col
    const int half = lane >> 4;                  // K sub-select (A/B), M-half (D)
    const int obase = wv * 16;
    const float invy = 1.0f / scal[0];
    const float* wr = wn + (long)(obase + nl) * KD;
    const float* bp = &Bt[nl * KD];

    v8f acc = {};
#pragma unroll 12
    for (int kc = 0; kc < KD / 4; ++kc) {        // 72 chunks of K=4
        const int k0 = kc * 4 + half * 2;
        const v2f a = *(const v2f*)(wr + k0);    // global_load_b64
        const v2f b = *(const v2f*)(bp + k0);    // ds_load_b64
        acc = __builtin_amdgcn_wmma_f32_16x16x4_f32(false, a, false, b, (short)0, acc,
                                                    false, false);
    }
#pragma unroll
    for (int r = 0; r < 8; ++r) {
        const int och = obase + r + half * 8;
        Xc[((long)bt * Dd + och) * HW + oy * Ww + ox0 + nl] = acc[r] * invy;
    }
}

// Transpose conv: xp[b,ci,y,x] = sum_{o,ky,kx} c[b,o,y+ky-1,x+kx-1]*wn[o,ci,2-ky,2-kx]
// Same structure, K=1152, M=32 (2 waves). Slab = 72KB LDS.
__global__ void conv_t_wmma(const float* __restrict__ c, const float* __restrict__ wtr,
                            float* __restrict__ xp) {
    __shared__ float Bt[16 * KDT];               // [col][k]  72KB
    const int ptile = blockIdx.x;
    const long pos0 = (long)ptile * 16;
    const int  bt  = (int)(pos0 / HW);
    const int  rm  = (int)(pos0 % HW);
    const int  oy  = rm / Ww;
    const int  ox0 = rm % Ww;
    const float* cb = c + (long)bt * Dd * HW;

    for (int idx = threadIdx.x; idx < 16 * KDT; idx += 64) {   // 288 iters exactly
        const int col = idx / KDT;
        const int k   = idx % KDT;
        const int o = k / 9, r9 = k % 9, ky = r9 / 3, kx = r9 % 3;
        const int iy = oy + ky - 1, ix = ox0 + col + kx - 1;
        float v = 0.f;
        if (iy >= 0 && iy < Hh && ix >= 0 && ix < Ww) v = cb[(long)o * HW + iy * Ww + ix];
        Bt[idx] = v;
    }
    __syncthreads();

    const int lane = threadIdx.x & 31;
    const int wv   = threadIdx.x >> 5;           // 0..1
    const int nl   = lane & 15;
    const int half = lane >> 4;
    const int cbase = wv * 16;
    const float* ar = wtr + (long)(cbase + nl) * KDT;
    const float* bp = &Bt[nl * KDT];

    v8f acc = {};
#pragma unroll 12
    for (int kc = 0; kc < KDT / 4; ++kc) {       // 288 chunks of K=4
        const int k0 = kc * 4 + half * 2;
        const v2f a = *(const v2f*)(ar + k0);
        const v2f b = *(const v2f*)(bp + k0);
        acc = __builtin_amdgcn_wmma_f32_16x16x4_f32(false, a, false, b, (short)0, acc,
                                                    false, false);
    }
#pragma unroll
    for (int r = 0; r < 8; ++r) {
        const int ci = cbase + r + half * 8;
        xp[((long)bt * Cin + ci) * HW + oy * Ww + ox0 + nl] = acc[r];
    }
}

// ---------------------------------------------------------------- power method
__global__ void x2_k(const float* __restrict__ Xc, const float* __restrict__ c,
                     float* __restrict__ X2) {
    long s = (long)gridDim.x * blockDim.x;
    for (long e = (long)blockIdx.x * blockDim.x + threadIdx.x; e < Nt; e += s)
        X2[e] = Xc[e] - c[e];
}

// all 4 mode-marginal sums of X2 in one pass  (y0 before normalization)
__global__ void init_marginals_k(const float* __restrict__ X2, float* __restrict__ ynew) {
    __shared__ float marg[328];                  // [8 | 128 | 96 | 96]
    for (int i = threadIdx.x; i < 328; i += 256) marg[i] = 0.f;
    __syncthreads();
    const long s = (long)gridDim.x * blockDim.x;     // divides Nt exactly
    for (long e = (long)blockIdx.x * blockDim.x + threadIdx.x; e < Nt; e += s) {
        const float v = X2[e];
        int i3 = (int)(e % 96); long t = e / 96;
        int i2 = (int)(t % 96); t /= 96;
        int i1 = (int)(t % 128);
        int i0 = (int)(t / 128);
        // i0,i1 are wave-uniform (strides are multiples of 32): reduce then one atomic
        float v0 = wave_sum(v);
        if ((threadIdx.x & 31) == 0) { atomicAdd(&marg[i0], v0); atomicAdd(&marg[8 + i1], v0); }
        atomicAdd(&marg[136 + i2], v);
        atomicAdd(&marg[232 + i3], v);
    }
    __syncthreads();
    for (int i = threadIdx.x; i < 8;   i += 256) atomicAdd(&ynew[0 * 128 + i], marg[i]);
    for (int i = threadIdx.x; i < 128; i += 256) atomicAdd(&ynew[1 * 128 + i], marg[8 + i]);
    for (int i = threadIdx.x; i < 96;  i += 256) atomicAdd(&ynew[2 * 128 + i], marg[136 + i]);
    for (int i = threadIdx.x; i < 96;  i += 256) atomicAdd(&ynew[3 * 128 + i], marg[232 + i]);
}

__global__ void normalize_y_k(const float* __restrict__ ynew, float* __restrict__ y) {
    __shared__ float red[4];
    const int d = blockIdx.x, t = threadIdx.x;
    const float v = ynew[d * 128 + t];
    float s = wave_sum(v * v);
    if ((t & 31) == 0) red[t >> 5] = s;
    __syncthreads();
    const float ss = red[0] + red[1] + red[2] + red[3];
    if (ss > 0.f) y[d * 128 + t] = v / sqrtf(ss);   // ny==0 -> keep y (reference where())
}

// fused  y' = Xm * (Xm^T * y)  — one pass over X2 (twice-read per column, L2 resident)
__global__ void power_pass_k(const float* __restrict__ X2, const float* __restrict__ y,
                             float* __restrict__ ynew, const int* __restrict__ maxd,
                             int mode_sel) {
    const int  m     = (mode_sel >= 0) ? mode_sel : *maxd;
    const int  n     = mode_n(m);
    const long str   = mode_stride(m);
    const long mcols = Nt / n;
    __shared__ float ys[128];
    __shared__ float yp[128];
    for (int i = threadIdx.x; i < 128; i += 256) { ys[i] = y[m * 128 + i]; yp[i] = 0.f; }
    __syncthreads();
    const long j   = (long)blockIdx.x * 256 + threadIdx.x;
    const bool act = (j < mcols);
    const long base = act ? col_base(m, j) : 0;
    float z = 0.f;
    if (act) for (int i = 0; i < n; ++i) z += X2[base + i * str] * ys[i];
    for (int i = 0; i < n; ++i) {
        float v = act ? X2[base + i * str] * z : 0.f;
        v = wave_sum(v);
        if ((threadIdx.x & 31) == 0) atomicAdd(&yp[i], v);
    }
    __syncthreads();
    for (int i = threadIdx.x; i < n; i += 256) atomicAdd(&ynew[m * 128 + i], yp[i]);
}

// z_j = (Xm^T y)_j ; accumulate ||z||^2 into sigslot; optionally write z into bout
__global__ void colpass_k(const float* __restrict__ X2, const float* __restrict__ y,
                          const int* __restrict__ maxd, int mode_sel,
                          float* __restrict__ bout, float* __restrict__ sigslot) {
    const int  m     = (mode_sel >= 0) ? mode_sel : *maxd;
    const int  n     = mode_n(m);
    const long str   = mode_stride(m);
    const long mcols = Nt / n;
    __shared__ float ys[128];
    __shared__ float part[8];
    for (int i = threadIdx.x; i < 128; i += 256) ys[i] = y[m * 128 + i];
    __syncthreads();
    const long j   = (long)blockIdx.x * 256 + threadIdx.x;
    const bool act = (j < mcols);
    float z = 0.f;
    if (act) {
        const long base = col_base(m, j);
        for (int i = 0; i < n; ++i) z += X2[base + i * str] * ys[i];
        if (bout != nullptr) bout[j] = z;
    }
    float v = wave_sum(act ? z * z : 0.f);
    if ((threadIdx.x & 31) == 0) part[threadIdx.x >> 5] = v;
    __syncthreads();
    if (threadIdx.x == 0) {
        float s = 0.f;
        for (int i = 0; i < 8; ++i) s += part[i];
        atomicAdd(sigslot, s);
    }
}

__global__ void argmax_k(const float* __restrict__ scal, int* __restrict__ maxd) {
    const float fac[4] = {2.828427124746190f, 11.31370849898476f,
                          9.797958971132712f, 9.797958971132712f};
    float best = -1.f; int bi = 0;
    for (int d = 0; d < 4; ++d) {
        const float s = sqrtf(scal[1 + d]) * fac[d];
        if (s > best) { best = s; bi = d; }       // first max wins (jnp.argmax)
    }
    *maxd = bi;
}

__global__ void scale_k(float* __restrict__ scal, const int* __restrict__ maxd) {
    const float fac[4] = {2.828427124746190f, 11.31370849898476f,
                          9.797958971132712f, 9.797958971132712f};
    const float sg = sqrtf(scal[5]);
    float sc = TAUc * fac[*maxd];
    if (sg != 0.f) sc /= sg;                     // v = b/sigma unless sigma==0
    scal[6] = sc; scal[7] = 0.f; scal[8] = 0.f;
}

__global__ void ls_pass_k(const float* __restrict__ X2, const float* __restrict__ c,
                          const float* __restrict__ bb, const float* __restrict__ y,
                          float* __restrict__ scal, const int* __restrict__ maxd) {
    const int m = *maxd;
    const float scale = scal[6];
    __shared__ float ys[128];
    __shared__ float p1[8], p2[8];
    for (int i = threadIdx.x; i < 128; i += 256) ys[i] = y[m * 128 + i];
    __syncthreads();
    float s1 = 0.f, s2 = 0.f;
    const long s = (long)gridDim.x * blockDim.x;
    for (long e = (long)blockIdx.x * blockDim.x + threadIdx.x; e < Nt; e += s) {
        int i; long j; elem_ij(m, e, i, j);
        const float nc  = scale * ys[i] * bb[j];
        const float cpx = nc - c[e];
        s1 += X2[e] * cpx;
        s2 += cpx * cpx;
    }
    s1 = wave_sum(s1); s2 = wave_sum(s2);
    if ((threadIdx.x & 31) == 0) { p1[threadIdx.x >> 5] = s1; p2[threadIdx.x >> 5] = s2; }
    __syncthreads();
    if (threadIdx.x == 0) {
        float a = 0.f, b = 0.f;
        for (int i = 0; i < 8; ++i) { a += p1[i]; b += p2[i]; }
        atomicAdd(&scal[7], a);
        atomicAdd(&scal[8], b);
    }
}

__global__ void gamma_k(float* __restrict__ scal) {
    float g = scal[7] / scal[8];
    g = fminf(fmaxf(g, 0.f), 1.f);
    scal[9] = (g == 0.f) ? 1e-6f : g;
}

__global__ void update_k(float* __restrict__ c, const float* __restrict__ bb,
                         const float* __restrict__ y, const float* __restrict__ scal,
                         const int* __restrict__ maxd) {
    const int m = *maxd;
    const float scale = scal[6];
    const float g = scal[9];
    __shared__ float ys[128];
    for (int i = threadIdx.x; i < 128; i += 256) ys[i] = y[m * 128 + i];
    __syncthreads();
    const long s = (long)gridDim.x * blockDim.x;
    for (long e = (long)blockIdx.x * blockDim.x + threadIdx.x; e < Nt; e += s) {
        int i; long j; elem_ij(m, e, i, j);
        const float nc = scale * ys[i] * bb[j];
        c[e] = (1.f - g) * c[e] + g * nc;
    }
}

__global__ void rloss_k(const float* __restrict__ x, const float* __restrict__ xp,
                        float* __restrict__ out) {
    __shared__ float part[8];
    float s = 0.f;
    const long st = (long)gridDim.x * blockDim.x;
    for (long e = (long)blockIdx.x * blockDim.x + threadIdx.x; e < Nx; e += st) {
        const float d = x[e] - xp[e];
        s += d * d;
    }
    s = wave_sum(s);
    if ((threadIdx.x & 31) == 0) part[threadIdx.x >> 5] = s;
    __syncthreads();
    if (threadIdx.x == 0) {
        float a = 0.f;
        for (int i = 0; i < 8; ++i) a += part[i];
        atomicAdd(out, a);
    }
}

__global__ void closs_k(const float* __restrict__ c, float* __restrict__ out) {
    __shared__ float part[8];
    float s = 0.f;
    const long st = (long)gridDim.x * blockDim.x;
    for (long e = (long)blockIdx.x * blockDim.x + threadIdx.x; e < Nt; e += st)
        s += fabsf(c[e]);
    s = wave_sum(s);
    if ((threadIdx.x & 31) == 0) part[threadIdx.x >> 5] = s;
    __syncthreads();
    if (threadIdx.x == 0) {
        float a = 0.f;
        for (int i = 0; i < 8; ++i) a += part[i];
        atomicAdd(out, a);
    }
}

// ---------------------------------------------------------------- launcher
extern "C" void kernel_launch(void* const* d_in, const int* in_sizes, int n_in,
                              void* d_out, int out_size, void* d_ws, size_t ws_size,
                              hipStream_t stream) {
    (void)in_sizes; (void)n_in; (void)out_size; (void)ws_size;
    const float* x = (const float*)d_in[0];
    const float* w = (const float*)d_in[1];
    float* out = (float*)d_out;           // [ c (Nt) | r_loss | c_loss ]
    float* ws  = (float*)d_ws;

    float* Xc   = ws;                     // Nt
    float* X2   = Xc + Nt;                // Nt
    float* bbuf = X2 + Nt;                // MMAX
    float* xp   = bbuf + MMAX;            // Nx
    float* wn   = xp + Nx;                // 128*288
    float* wtr  = wn + Dd * KD;           // 32*1152 (reordered transpose weights)
    float* ybuf = wtr + Cin * KDT;        // 4*128
    float* ynew = ybuf + 512;             // 4*128
    float* scal = ynew + 512;             // 16 scalars
    int*   maxd = (int*)(scal + 16);

    float* cP = out;

    const dim3 b256(256);
    const int gpw[4] = {4608, 288, 384, 384};   // exact column grids per mode

    zero_k<<<4096, b256, 0, stream>>>(out, Nt + 2);
    normalize_weight_k<<<Dd, 96, 0, stream>>>(w, wn);
    yita_k<<<1, b256, 0, stream>>>(wn, scal);
    reorder_wt_k<<<16, b256, 0, stream>>>(wn, wtr);
    conv_fwd_wmma<<<4608, b256, 0, stream>>>(x, wn, scal, Xc);

    for (int t = 0; t < 10; ++t) {
        x2_k<<<4096, b256, 0, stream>>>(Xc, cP, X2);

        // init y0 for all 4 modes: normalized marginal sums
        zero_k<<<2, b256, 0, stream>>>(ynew, 512);
        init_marginals_k<<<1024, b256, 0, stream>>>(X2, ynew);
        normalize_y_k<<<4, 128, 0, stream>>>(ynew, ybuf);

        // first stage: 3 fused power iterations, each of the 4 modes
        for (int it = 0; it < 3; ++it) {
            zero_k<<<2, b256, 0, stream>>>(ynew, 512);
            for (int m = 0; m < 4; ++m)
                power_pass_k<<<gpw[m], b256, 0, stream>>>(X2, ybuf, ynew, maxd, m);
            normalize_y_k<<<4, 128, 0, stream>>>(ynew, ybuf);
        }

        // sigma per mode, argmax (device-side; later kernels are predicated on maxd)
        zero_k<<<1, 32, 0, stream>>>(scal + 1, 9);
        for (int m = 0; m < 4; ++m)
            colpass_k<<<gpw[m], b256, 0, stream>>>(X2, ybuf, maxd, m, nullptr, scal + 1 + m);
        argmax_k<<<1, 1, 0, stream>>>(scal, maxd);

        // second stage: 8 more iterations on the winning mode only
        for (int it = 0; it < 8; ++it) {
            zero_k<<<2, b256, 0, stream>>>(ynew, 512);
            power_pass_k<<<4608, b256, 0, stream>>>(X2, ybuf, ynew, maxd, -1);
            normalize_y_k<<<4, 128, 0, stream>>>(ynew, ybuf);
        }

        // b = Xm^T u, sigma = ||b||; newcomp(e) = (TAU*fac/sigma)*u[i]*b[j]
        colpass_k<<<4608, b256, 0, stream>>>(X2, ybuf, maxd, -1, bbuf, scal + 5);
        scale_k<<<1, 1, 0, stream>>>(scal, maxd);
        ls_pass_k<<<4096, b256, 0, stream>>>(X2, cP, bbuf, ybuf, scal, maxd);
        gamma_k<<<1, 1, 0, stream>>>(scal);
        update_k<<<4096, b256, 0, stream>>>(cP, bbuf, ybuf, scal, maxd);
    }

    conv_t_wmma<<<4608, 64, 0, stream>>>(cP, wtr, xp);
    rloss_k<<<1024, b256, 0, stream>>>(x, xp, out + Nt);
    closs_k<<<4096, b256, 0, stream>>>(cP, out + Nt + 1);
}